// BettingLoss_50603304682293
// MI455X (gfx1250) — compile-verified
//
#include <hip/hip_runtime.h>

typedef __attribute__((ext_vector_type(2))) float v2f;
typedef __attribute__((ext_vector_type(8))) float v8f;

#define BATCH   (1 << 20)     // 1048576 races
#define NDOG    8
#define TPB     256           // 8 waves per block (wave32)
#define NBLK    1024
#define GSTRIDE (TPB * NBLK)  // 262144 threads
#define RPT     (BATCH / GSTRIDE)  // 4 rows per thread

// Fast hardware transcendentals: v_exp_f32 / v_log_f32 / v_rcp_f32 (~1 ulp).
// All consumers are means over ~1e6 rows; ulp-level error is invisible.
__device__ __forceinline__ float fexp(float x) { return __expf(x); }
__device__ __forceinline__ float flog(float x) { return __logf(x); }
__device__ __forceinline__ float frcp(float x) { return __builtin_amdgcn_rcpf(x); }

// ---------------------------------------------------------------------------
// Kernel 1: streaming per-row math + hierarchical reduction to per-block
// partials. ws layout: NBLK rows x 16 floats (cols 0..5 = partials, 6..15 = 0)
//   q0=cnt  q1=sum(ce*valid)  q2=sum(ce)  q3=sum(sep*valid)
//   q4=sum(max p)  q5=sum(entropy)
// Roofline: 128 MB read / 23.3 TB/s ~= 5.5 us floor (or L2-resident on
// replay: 128 MB < 192 MB L2). Hardware TRANS ops keep dynamic VALU at
// ~150 lane-ops/row << memory time, so we stay on the bandwidth roof.
// ---------------------------------------------------------------------------
__global__ __launch_bounds__(TPB) void bl_partials(
    const float* __restrict__ probs, const float* __restrict__ winners,
    const float* __restrict__ odds,  const float* __restrict__ gumbel,
    float* __restrict__ ws)
{
  const int tid = threadIdx.x;
  const int gid = blockIdx.x * TPB + tid;

  float a_cnt = 0.f, a_cev = 0.f, a_cet = 0.f;
  float a_sep = 0.f, a_mxp = 0.f, a_ent = 0.f;

  #pragma unroll
  for (int it = 0; it < RPT; ++it) {
    const size_t r = (size_t)gid + (size_t)it * GSTRIDE;

    const float4* p4 = (const float4*)(probs   + r * NDOG);
    const float4* w4 = (const float4*)(winners + r * NDOG);
    const float4* o4 = (const float4*)(odds    + r * NDOG);
    const float4* g4 = (const float4*)(gumbel  + r * NDOG);
    float4 pA = p4[0], pB = p4[1];
    float4 wA = w4[0], wB = w4[1];
    float4 oA = o4[0], oB = o4[1];
    float4 gA = g4[0], gB = g4[1];

    if (it + 1 < RPT) {   // gfx1250 global_prefetch_b8: pull next strided tile
      const size_t rn = (r + (size_t)GSTRIDE) * NDOG;
      __builtin_prefetch(probs   + rn, 0, 1);
      __builtin_prefetch(winners + rn, 0, 1);
      __builtin_prefetch(odds    + rn, 0, 1);
      __builtin_prefetch(gumbel  + rn, 0, 1);
    }

    float p[8] = {pA.x, pA.y, pA.z, pA.w, pB.x, pB.y, pB.z, pB.w};
    float w[8] = {wA.x, wA.y, wA.z, wA.w, wB.x, wB.y, wB.z, wB.w};
    float o[8] = {oA.x, oA.y, oA.z, oA.w, oB.x, oB.y, oB.z, oB.w};
    float g[8] = {gA.x, gA.y, gA.z, gA.w, gB.x, gB.y, gB.z, gB.w};

    // --- race validity: any(odds>0) && sum(1/max(odds,1.01)) >= 0.95 ---
    bool  anyodds = false;
    float simp    = 0.f;
    #pragma unroll
    for (int j = 0; j < 8; ++j) {
      anyodds = anyodds || (o[j] > 0.f);
      simp   += frcp(fmaxf(o[j], 1.01f));   // v_rcp_f32: threshold flip risk
    }                                       // is O(1 row in 1e6), ~2e-6 rel.
    const bool  valid = anyodds && (simp >= 0.95f);
    const float vf    = valid ? 1.0f : 0.0f;

    // --- masked expected profit (reference association) ---
    float ep[8];
    #pragma unroll
    for (int j = 0; j < 8; ++j) {
      const float e = (o[j] * 1.1f * p[j] - 1.0f) * 0.02f * 0.95f;
      ep[j] = valid ? e : 0.0f;
    }

    // --- gumbel-softmax selection: softmax((ep/0.1 + g)/0.1) ---
    // /0.1 twice folded to *100 / *10 (softmax is shift/ulp insensitive)
    float l[8];
    float lmax = -__builtin_inff();
    #pragma unroll
    for (int j = 0; j < 8; ++j) {
      l[j] = ep[j] * 100.0f + g[j] * 10.0f;
      lmax = fmaxf(lmax, l[j]);
    }
    float ssum = 0.f, snum = 0.f;
    #pragma unroll
    for (int j = 0; j < 8; ++j) {
      const float e = fexp(l[j] - lmax);
      ssum += e;
      snum += e * ep[j];
    }
    const float sep = snum * frcp(ssum);

    // --- CE with probs-as-logits: lse(p) - dot(onehot, p); max(p); entropy ---
    float pmax = -__builtin_inff();
    #pragma unroll
    for (int j = 0; j < 8; ++j) pmax = fmaxf(pmax, p[j]);
    float esum = 0.f, dot = 0.f, ent = 0.f;
    #pragma unroll
    for (int j = 0; j < 8; ++j) {
      esum += fexp(p[j] - pmax);
      dot  += w[j] * p[j];
      ent  -= p[j] * flog(p[j] + 1e-8f);
    }
    const float ce = (pmax + flog(esum)) - dot;

    a_cnt += vf;       a_cev += vf * ce;  a_cet += ce;
    a_sep += vf * sep; a_mxp += pmax;     a_ent += ent;
  }

  // --- wave32 shuffle tree ---
  #pragma unroll
  for (int off = 16; off > 0; off >>= 1) {
    a_cnt += __shfl_down(a_cnt, off, 32);
    a_cev += __shfl_down(a_cev, off, 32);
    a_cet += __shfl_down(a_cet, off, 32);
    a_sep += __shfl_down(a_sep, off, 32);
    a_mxp += __shfl_down(a_mxp, off, 32);
    a_ent += __shfl_down(a_ent, off, 32);
  }

  __shared__ float sm[8][6];
  const int wave = tid >> 5, lane = tid & 31;
  if (lane == 0) {
    sm[wave][0] = a_cnt; sm[wave][1] = a_cev; sm[wave][2] = a_cet;
    sm[wave][3] = a_sep; sm[wave][4] = a_mxp; sm[wave][5] = a_ent;
  }
  __syncthreads();
  // write padded 16-wide partial row (cols 6..15 zero -> free WMMA padding)
  if (tid < 16) {
    float s = 0.f;
    if (tid < 6) {
      #pragma unroll
      for (int wv = 0; wv < 8; ++wv) s += sm[wv][tid];
    }
    ws[blockIdx.x * 16 + tid] = s;
  }
}

// ---------------------------------------------------------------------------
// Kernel 2: column-sum of the 1024x16 partial matrix via the matrix pipe:
// D = ones(16x4) x B(4x16) + C accumulated over chunks of 4 rows.
// A is all-ones, so D[m,n] = sum_k B[k,n] under any bijective K-slot layout.
// 8 waves x 4 interleaved f32-WMMA accumulator chains (bit-equivalent to
// scalar f32 adds; EXEC all-ones through the loop as WMMA requires).
// ---------------------------------------------------------------------------
__device__ __forceinline__ v2f load_bcols(const float* __restrict__ ws,
                                          int chunk, int half, int col) {
  const int base = chunk << 2;   // 4 partial-rows per chunk
  v2f b;
  b.x = ws[(base + half) * 16 + col];       // k-slot 0/1
  b.y = ws[(base + 2 + half) * 16 + col];   // k-slot 2/3
  return b;
}

__global__ __launch_bounds__(TPB) void bl_final(const float* __restrict__ ws,
                                                float* __restrict__ out)
{
  const int tid  = threadIdx.x;
  const int wave = tid >> 5;
  const int lane = tid & 31;
  const int half = lane >> 4;
  const int col  = lane & 15;

  const v2f a = {1.0f, 1.0f};          // ones(16x4): every A slot = 1
  v8f c0 = (v8f)0.f, c1 = (v8f)0.f, c2 = (v8f)0.f, c3 = (v8f)0.f;

  // NBLK/4 = 256 chunks; wave handles chunks wave, wave+8, ... (32 each)
  #pragma unroll 2
  for (int i = 0; i < 32; i += 4) {
    const v2f b0 = load_bcols(ws, wave + ((i + 0) << 3), half, col);
    const v2f b1 = load_bcols(ws, wave + ((i + 1) << 3), half, col);
    const v2f b2 = load_bcols(ws, wave + ((i + 2) << 3), half, col);
    const v2f b3 = load_bcols(ws, wave + ((i + 3) << 3), half, col);
    c0 = __builtin_amdgcn_wmma_f32_16x16x4_f32(false, a, false, b0, (short)0, c0, false, false);
    c1 = __builtin_amdgcn_wmma_f32_16x16x4_f32(false, a, false, b1, (short)0, c1, false, false);
    c2 = __builtin_amdgcn_wmma_f32_16x16x4_f32(false, a, false, b2, (short)0, c2, false, false);
    c3 = __builtin_amdgcn_wmma_f32_16x16x4_f32(false, a, false, b3, (short)0, c3, false, false);
  }
  const v8f ct = c0 + c1 + c2 + c3;
  // D layout: ct[0] on lanes 0-15 is (M=0, N=lane); all M rows identical.
  const float mysum = ct[0];

  __shared__ float sm[8][6];
  if (lane < 6) sm[wave][lane] = mysum;
  __syncthreads();

  if (tid == 0) {
    float tot[6];
    #pragma unroll
    for (int q = 0; q < 6; ++q) {
      float s = 0.f;
      #pragma unroll
      for (int wv = 0; wv < 8; ++wv) s += sm[wv][q];
      tot[q] = s;
    }
    const float cnt = tot[0], cev = tot[1], cet = tot[2];
    const float sep = tot[3], mxp = tot[4], ent = tot[5];
    const float Bf  = (float)BATCH;

    const float pred = (cnt > 0.f) ? (cev / fmaxf(cnt, 1.0f)) : (cet / Bf);
    const float conf = -(mxp / Bf) * 0.1f;
    const float bet  = (cnt > 0.f) ? (-sep / Bf) : conf;
    const float entr = ent / Bf;
    const float lam  = fminf(0.5f + cnt / 10000.0f * 0.5f, 1.0f);
    out[0] = pred + lam * bet - 0.01f * entr;
  }
}

// ---------------------------------------------------------------------------
extern "C" void kernel_launch(void* const* d_in, const int* in_sizes, int n_in,
                              void* d_out, int out_size, void* d_ws, size_t ws_size,
                              hipStream_t stream) {
  (void)in_sizes; (void)n_in; (void)out_size; (void)ws_size;
  const float* probs   = (const float*)d_in[0];
  const float* winners = (const float*)d_in[1];
  const float* odds    = (const float*)d_in[2];
  const float* gumbel  = (const float*)d_in[3];
  float* ws  = (float*)d_ws;   // needs NBLK*16*4 = 64 KB
  float* out = (float*)d_out;

  bl_partials<<<NBLK, TPB, 0, stream>>>(probs, winners, odds, gumbel, ws);
  bl_final<<<1, TPB, 0, stream>>>(ws, out);
}